// GraphPoolLayer_72404558676396
// MI455X (gfx1250) — compile-verified
//
#include <hip/hip_runtime.h>
#include <math.h>

typedef __attribute__((ext_vector_type(2))) float v2f;
typedef __attribute__((ext_vector_type(8))) float v8f;

#define BB 32
#define NN 16384
#define FF 256
#define KK 8192   // top-k = N/2

// ---------------------------------------------------------------------------
// Pass 1: scores[b,n] = dot(inputs[b,n,:], proj[:]) via V_WMMA_F32_16X16X4_F32
// One wave computes 16 scores (16 rows of the flattened [B*N, F] matrix).
// A fragment (16x4 f32, ISA layout): lanes 0-15 hold M=0..15 K={0,1},
// lanes 16-31 hold M=0..15 K={2,3}  -> one b64 load per lane per K-chunk.
// B fragment (4x16 f32): proj[k..k+3] broadcast across all 16 N columns, so
// every column of D is the score vector; lanes 0 and 16 write it out.
// ---------------------------------------------------------------------------
__global__ __launch_bounds__(256) void score_wmma_kernel(
    const float* __restrict__ inputs, const float* __restrict__ proj,
    float* __restrict__ scores)
{
    __shared__ float sproj[FF];
    for (int i = threadIdx.x; i < FF; i += 256) sproj[i] = proj[i];
    __syncthreads();

    const int lane  = threadIdx.x & 31;
    const int wid   = (blockIdx.x << 3) + (threadIdx.x >> 5); // 16-row group id
    const int mrow  = lane & 15;
    const int khalf = (lane >> 4) << 1;                       // 0 or 2

    const float* aptr = inputs + ((long)wid * 16 + mrow) * FF + khalf;

    v8f acc = {};
    #pragma unroll 4
    for (int k = 0; k < FF; k += 4) {
        v2f a = *(const v2f*)(aptr + k);      // A: elements k+khalf, k+khalf+1
        v2f b;
        b.x = sproj[k + khalf];               // B row K=khalf   (broadcast over N)
        b.y = sproj[k + khalf + 1];           // B row K=khalf+1
        acc = __builtin_amdgcn_wmma_f32_16x16x4_f32(
            /*neg_a=*/false, a, /*neg_b=*/false, b,
            /*c_mod=*/(short)0, acc, /*reuse_a=*/false, /*reuse_b=*/false);
    }

    // D layout: VGPR v, lanes 0-15 -> M=v ; lanes 16-31 -> M=v+8 (any N column)
    if ((lane & 15) == 0) {
        float* sp = scores + (long)wid * 16 + (lane >> 4) * 8;
        #pragma unroll
        for (int v = 0; v < 8; ++v) sp[v] = acc[v];
    }
}

// ---------------------------------------------------------------------------
// Pass 2: exact top_k (sorted, stable) per batch row.
// One 1024-thread block per row. Full row (key,val) double-buffered in LDS
// (256 KB) -- possible because gfx1250 has 320 KB LDS per WGP.
// Stable LSD radix sort, 4 x 8-bit passes, on descending-transformed keys:
//   asc = bits ^ (sign ? 0xFFFFFFFF : 0x80000000);  key = ~asc
// ascending sort of `key` == descending score, stability == lower-index ties.
// ---------------------------------------------------------------------------
__global__ __launch_bounds__(1024) void topk_sort_kernel(
    const float* __restrict__ scores, unsigned* __restrict__ sorted_idx)
{
    extern __shared__ unsigned smem[];
    unsigned* keys0 = smem;                 // NN
    unsigned* vals0 = keys0 + NN;           // NN
    unsigned* keys1 = vals0 + NN;           // NN
    unsigned* vals1 = keys1 + NN;           // NN
    unsigned* hist  = vals1 + NN;           // 256 * 32  (digit-major)
    unsigned* dbase = hist + 256 * 32;      // 256

    const int tid  = threadIdx.x;
    const int w    = tid >> 5;
    const int lane = tid & 31;
    const int brow = blockIdx.x;
    const float* srow = scores + (long)brow * NN;

    for (int i = tid; i < NN; i += 1024) {
        unsigned u   = __float_as_uint(srow[i]);
        unsigned asc = u ^ ((u & 0x80000000u) ? 0xFFFFFFFFu : 0x80000000u);
        keys0[i] = ~asc;
        vals0[i] = (unsigned)i;
    }
    __syncthreads();

    unsigned* ik = keys0; unsigned* iv = vals0;
    unsigned* ok = keys1; unsigned* ov = vals1;

    for (int p = 0; p < 4; ++p) {
        const int shift = p * 8;

        for (int i = tid; i < 256 * 32; i += 1024) hist[i] = 0;
        __syncthreads();

        // per-wave digit histogram over this wave's contiguous 512 elements
        for (int r = 0; r < 16; ++r) {
            unsigned key = ik[w * 512 + r * 32 + lane];
            unsigned d = (key >> shift) & 255u;
            atomicAdd(&hist[d * 32 + w], 1u);
        }
        __syncthreads();

        // exclusive scan across waves within each digit; totals -> dbase
        if (tid < 256) {
            unsigned run = 0;
            for (int ww = 0; ww < 32; ++ww) {
                unsigned c = hist[tid * 32 + ww];
                hist[tid * 32 + ww] = run;
                run += c;
            }
            dbase[tid] = run;
        }
        __syncthreads();
        if (tid == 0) {             // exclusive scan over 256 digit totals
            unsigned run = 0;
            for (int d = 0; d < 256; ++d) {
                unsigned c = dbase[d];
                dbase[d] = run;
                run += c;
            }
        }
        __syncthreads();

        // stable scatter: rounds in element order; wave32 multisplit per round
        for (int r = 0; r < 16; ++r) {
            const int i = w * 512 + r * 32 + lane;
            unsigned key = ik[i];
            unsigned val = iv[i];
            unsigned d = (key >> shift) & 255u;

            unsigned mask = 0xFFFFFFFFu;           // match-any over 8-bit digit
            #pragma unroll
            for (int bit = 0; bit < 8; ++bit) {
                unsigned bal = (unsigned)__ballot((d >> bit) & 1u);
                mask &= ((d >> bit) & 1u) ? bal : ~bal;
            }
            int      leader = __ffs(mask) - 1;
            unsigned prefix = __popc(mask & ((1u << lane) - 1u));
            unsigned cnt    = __popc(mask);

            unsigned old = 0;
            if (lane == leader) old = atomicAdd(&hist[d * 32 + w], cnt);
            unsigned gbase = (unsigned)__shfl((int)old, leader);

            unsigned pos = dbase[d] + gbase + prefix;
            ok[pos] = key;
            ov[pos] = val;
        }
        __syncthreads();

        unsigned* t;
        t = ik; ik = ok; ok = t;
        t = iv; iv = ov; ov = t;
    }

    // 4 passes -> result back in keys0/vals0 (== ik/iv); emit top-K indices
    unsigned* outbase = sorted_idx + (long)brow * NN;
    for (int i = tid; i < KK; i += 1024) outbase[i] = iv[i];
}

// ---------------------------------------------------------------------------
// Pass 3: out[b,j,:] = inputs[b, idx[b,j], :] * sigmoid(scores[b, idx[b,j]])
// One 32-lane group per output row; float4 streaming (512B per group per step).
// ---------------------------------------------------------------------------
__global__ __launch_bounds__(256) void gather_gate_kernel(
    const float* __restrict__ inputs, const float* __restrict__ scores,
    const unsigned* __restrict__ sorted_idx, float* __restrict__ out)
{
    const int  lane = threadIdx.x & 31;
    const long orow = (long)blockIdx.x * 8 + (threadIdx.x >> 5);
    const int  b    = (int)(orow >> 13);     // / 8192
    const int  j    = (int)(orow & 8191);

    const unsigned idx = sorted_idx[(long)b * NN + j];
    const float    s   = scores[(long)b * NN + idx];
    const float    g   = 1.0f / (1.0f + __expf(-s));

    const float4* src = (const float4*)(inputs + ((long)b * NN + idx) * FF);
    float4*       dst = (float4*)(out + orow * FF);
    #pragma unroll
    for (int t = 0; t < 2; ++t) {
        float4 v = src[t * 32 + lane];
        v.x *= g; v.y *= g; v.z *= g; v.w *= g;
        dst[t * 32 + lane] = v;
    }
}

// ---------------------------------------------------------------------------
extern "C" void kernel_launch(void* const* d_in, const int* in_sizes, int n_in,
                              void* d_out, int out_size, void* d_ws, size_t ws_size,
                              hipStream_t stream)
{
    const float* inputs = (const float*)d_in[0];   // [32,16384,256] f32
    const float* proj   = (const float*)d_in[1];   // [256,1] f32
    float*       out    = (float*)d_out;           // [32,8192,256] f32

    float*    scores     = (float*)d_ws;                                   // 2 MB
    unsigned* sorted_idx = (unsigned*)((char*)d_ws +
                                       (size_t)BB * NN * sizeof(float));   // 2 MB

    // Pass 1: 524288 rows / 16 per wave / 8 waves per block = 4096 blocks
    score_wmma_kernel<<<(BB * NN / 16) / 8, 256, 0, stream>>>(inputs, proj, scores);

    // Pass 2: one block per batch row; dynamic LDS ~289 KB (fits 320 KB/WGP)
    size_t smem = (size_t)(4 * NN + 256 * 32 + 256) * sizeof(unsigned);
    topk_sort_kernel<<<BB, 1024, smem, stream>>>(scores, sorted_idx);

    // Pass 3: 262144 output rows / 8 per block = 32768 blocks
    gather_gate_kernel<<<(BB * KK) / 8, 256, 0, stream>>>(inputs, scores, sorted_idx, out);
}